// MDTA_90039694393992
// MI455X (gfx1250) — compile-verified
//
#include <hip/hip_runtime.h>
#include <math.h>

typedef float v2f __attribute__((ext_vector_type(2)));
typedef float v8f __attribute__((ext_vector_type(8)));

#define NB 8
#define CC 192
#define HH 128
#define WW2 128
#define NN (HH * WW2)        // 16384
#define NHEADS 4
#define HD 48
#define KK 192               // K dim of both 1x1-conv GEMMs
#define EPSN 1e-12f
#define CK 32                // K-chunk for attention GEMM

// ---------------------------------------------------------------------------
// GEMM: C[M,N] = W[M,K=192] * X[K=192,N=16384] per batch (z).
// Block = 256 thr = 8 waves laid out 2(M) x 4(N); block tile 32x64; wave 16x16.
// A panel (32 x 192) staged in LDS; B streamed from global (coalesced over n).
// WMMA: V_WMMA_F32_16X16X4_F32, K-loop of 48 steps.
// ---------------------------------------------------------------------------
__global__ __launch_bounds__(256)
void gemm_wmma(const float* __restrict__ Wm, long wStride,
               const float* __restrict__ Xm, long xStride,
               float* __restrict__ Cm, long cStride)
{
    __shared__ float ldsA[32 * KK];
    const int b = blockIdx.z;
    const float* Wp = Wm + (size_t)b * wStride;
    const float* Xp = Xm + (size_t)b * xStride;
    float*       Cp = Cm + (size_t)b * cStride;

    const int m0  = blockIdx.y * 32;
    const int n0  = blockIdx.x * 64;
    const int tid = threadIdx.x;

    for (int i = tid; i < 32 * KK; i += 256) {
        int r = i / KK, c = i - r * KK;
        ldsA[i] = Wp[(size_t)(m0 + r) * KK + c];
    }
    __syncthreads();

    const int wave = tid >> 5;
    const int lane = tid & 31;
    const int wy   = wave >> 2;      // 0..1  (M)
    const int wx   = wave & 3;       // 0..3  (N)
    const int l15  = lane & 15;
    const int hi   = lane >> 4;      // lanes 16-31 carry K+2,K+3
    const int mLoc = wy * 16 + l15;
    const int nCol = n0 + wx * 16 + l15;

    v8f acc = {0.f, 0.f, 0.f, 0.f, 0.f, 0.f, 0.f, 0.f};
    for (int k = 0; k < KK; k += 4) {
        // A 16x4 tile from LDS: lane<16 -> (k,k+1), lane>=16 -> (k+2,k+3)
        const float* ap = &ldsA[mLoc * KK + k + hi * 2];
        v2f a; a.x = ap[0]; a.y = ap[1];
        // B 4x16 tile from global: VGPR0 = row (k + 2*hi), VGPR1 = row (k+1+2*hi)
        v2f bb;
        bb.x = Xp[(size_t)(k + hi * 2) * NN + nCol];
        bb.y = Xp[(size_t)(k + 1 + hi * 2) * NN + nCol];
        if (k + 16 < KK)
            __builtin_prefetch(&Xp[(size_t)(k + 16 + hi * 2) * NN + nCol], 0, 1);
        acc = __builtin_amdgcn_wmma_f32_16x16x4_f32(false, a, false, bb,
                                                    (short)0, acc, false, false);
    }
#pragma unroll
    for (int j = 0; j < 8; ++j) {
        int row = m0 + wy * 16 + (hi ? (j + 8) : j);
        Cp[(size_t)row * NN + nCol] = acc[j];
    }
}

// ---------------------------------------------------------------------------
// Depthwise 3x3 conv (SAME, groups=3C) + fused sum-of-squares for q/k channels.
// ---------------------------------------------------------------------------
__global__ __launch_bounds__(256)
void dwconv_ssq(const float* __restrict__ qkv, const float* __restrict__ wdw,
                float* __restrict__ out, float* __restrict__ ssq)
{
    __shared__ float red[256];
    const int p  = blockIdx.x * 256 + threadIdx.x;
    const int ch = blockIdx.y;           // 0..575
    const int b  = blockIdx.z;
    const int y = p >> 7, x = p & (WW2 - 1);
    const float* in = qkv + ((size_t)b * 3 * CC + ch) * NN;

    float wv[9];
#pragma unroll
    for (int t = 0; t < 9; ++t) wv[t] = wdw[ch * 9 + t];

    float acc = 0.f;
#pragma unroll
    for (int ky = 0; ky < 3; ++ky) {
        int yy = y + ky - 1;
        if (yy < 0 || yy >= HH) continue;
#pragma unroll
        for (int kx = 0; kx < 3; ++kx) {
            int xx = x + kx - 1;
            if (xx < 0 || xx >= WW2) continue;
            acc += wv[ky * 3 + kx] * in[yy * WW2 + xx];
        }
    }
    out[((size_t)b * 3 * CC + ch) * NN + p] = acc;

    if (ch < 2 * CC) {  // q and k channels need L2 norms over N
        red[threadIdx.x] = acc * acc;
        __syncthreads();
        for (int s = 128; s > 0; s >>= 1) {
            if (threadIdx.x < s) red[threadIdx.x] += red[threadIdx.x + s];
            __syncthreads();
        }
        if (threadIdx.x == 0)
            atomicAdd(&ssq[(size_t)b * 2 * CC + ch], red[0]);
    }
}

// ---------------------------------------------------------------------------
// Channel attention: S = q*k^T over N=16384 (raw dots, WMMA f32 16x16x4),
// then scale by 1/max(||q_c||,eps) * 1/max(||k_d||,eps) * temperature, softmax.
// Block = 288 thr = 9 waves; wave w -> tile (mi=w/3, ni=w%3) of the 48x48 S.
// q/k K-chunks (48 x 32) staged in LDS -> A and B are contiguous float2 reads.
// ---------------------------------------------------------------------------
__global__ __launch_bounds__(288)
void attn_wmma(const float* __restrict__ qkv_dw, const float* __restrict__ ssq,
               const float* __restrict__ temperature, float* __restrict__ attn)
{
    __shared__ float ldsQ[HD * CK];
    __shared__ float ldsK[HD * CK];
    __shared__ float sbuf[HD][HD + 1];
    __shared__ float rk[HD];

    const int h = blockIdx.x, b = blockIdx.y;
    const float* qb = qkv_dw + ((size_t)b * 3 * CC + h * HD) * NN;
    const float* kb = qkv_dw + ((size_t)b * 3 * CC + CC + h * HD) * NN;

    const int tid  = threadIdx.x;
    const int wave = tid >> 5, lane = tid & 31;
    const int mi = wave / 3, ni = wave - mi * 3;
    const int l15 = lane & 15, hi = lane >> 4;

    v8f acc = {0.f, 0.f, 0.f, 0.f, 0.f, 0.f, 0.f, 0.f};
    for (int ko = 0; ko < NN; ko += CK) {
        __syncthreads();
        for (int i = tid; i < HD * CK; i += 288) {
            int r = i / CK, c = i - r * CK;
            ldsQ[i] = qb[(size_t)r * NN + ko + c];
            ldsK[i] = kb[(size_t)r * NN + ko + c];
        }
        __syncthreads();
#pragma unroll
        for (int kk = 0; kk < CK; kk += 4) {
            const float* ap = &ldsQ[(mi * 16 + l15) * CK + kk + hi * 2];
            const float* bp = &ldsK[(ni * 16 + l15) * CK + kk + hi * 2];
            v2f a;  a.x  = ap[0]; a.y  = ap[1];
            v2f bb; bb.x = bp[0]; bb.y = bp[1];
            acc = __builtin_amdgcn_wmma_f32_16x16x4_f32(false, a, false, bb,
                                                        (short)0, acc, false, false);
        }
    }

    __syncthreads();
#pragma unroll
    for (int j = 0; j < 8; ++j) {
        int row = mi * 16 + (hi ? (j + 8) : j);
        sbuf[row][ni * 16 + l15] = acc[j];
    }
    if (tid < HD)
        rk[tid] = 1.f / fmaxf(sqrtf(ssq[(size_t)b * 2 * CC + CC + h * HD + tid]), EPSN);
    __syncthreads();

    if (tid < HD) {
        float rq = 1.f / fmaxf(sqrtf(ssq[(size_t)b * 2 * CC + h * HD + tid]), EPSN);
        float t = temperature[h] * rq;
        float mx = -3.4e38f;
        for (int d = 0; d < HD; ++d) {
            float v = sbuf[tid][d] * t * rk[d];
            sbuf[tid][d] = v;
            mx = fmaxf(mx, v);
        }
        float s = 0.f;
        for (int d = 0; d < HD; ++d) {
            float e = __expf(sbuf[tid][d] - mx);
            sbuf[tid][d] = e;
            s += e;
        }
        float inv = 1.f / s;
        float* ap = attn + (((size_t)b * NHEADS + h) * HD + tid) * HD;
        for (int d = 0; d < HD; ++d) ap[d] = sbuf[tid][d] * inv;
    }
}

// ---------------------------------------------------------------------------
// F[b][o][h*48+d] = sum_cc w_out[o][h*48+cc] * attn[b][h][cc][d]
// (folds attn into the output projection; tiny: 28 MFLOP total)
// ---------------------------------------------------------------------------
__global__ __launch_bounds__(256)
void fuse_wout_attn(const float* __restrict__ wout, const float* __restrict__ attn,
                    float* __restrict__ F)
{
    int idx = blockIdx.x * 256 + threadIdx.x;      // o*192 + c
    int b = blockIdx.y;
    int o = idx / CC, c = idx - o * CC;
    int hh = c / HD, d = c - hh * HD;
    const float* wrow = wout + (size_t)o * CC + hh * HD;
    const float* arow = attn + ((size_t)b * NHEADS + hh) * HD * HD + d;
    float s = 0.f;
#pragma unroll
    for (int cc = 0; cc < HD; ++cc) s += wrow[cc] * arow[(size_t)cc * HD];
    F[((size_t)b * CC + o) * CC + c] = s;
}

__global__ __launch_bounds__(256)
void zero_f32(float* __restrict__ p, int n)
{
    int i = blockIdx.x * 256 + threadIdx.x;
    if (i < n) p[i] = 0.f;
}

// ---------------------------------------------------------------------------
extern "C" void kernel_launch(void* const* d_in, const int* in_sizes, int n_in,
                              void* d_out, int out_size, void* d_ws, size_t ws_size,
                              hipStream_t stream)
{
    const float* x      = (const float*)d_in[0];   // [8,192,128,128]
    const float* w_qkv  = (const float*)d_in[1];   // [576,192]
    const float* w_dw   = (const float*)d_in[2];   // [576,1,3,3]
    const float* temp   = (const float*)d_in[3];   // [1,4,1,1]
    const float* w_out  = (const float*)d_in[4];   // [192,192]
    float* out = (float*)d_out;

    float* ws     = (float*)d_ws;
    float* qkv    = ws;                                        // 8*576*16384
    float* qkvdw  = qkv   + (size_t)NB * 3 * CC * NN;          // 8*576*16384
    float* ssq    = qkvdw + (size_t)NB * 3 * CC * NN;          // 8*384
    float* attn   = ssq   + (size_t)NB * 2 * CC;               // 8*4*48*48
    float* F      = attn  + (size_t)NB * NHEADS * HD * HD;     // 8*192*192

    // 1) qkv = w_qkv (576x192, shared) x x_b (192x16384)
    gemm_wmma<<<dim3(NN / 64, (3 * CC) / 32, NB), 256, 0, stream>>>(
        w_qkv, 0L, x, (long)CC * NN, qkv, (long)3 * CC * NN);

    // 2) zero the sum-of-squares table (ws is poisoned each run)
    zero_f32<<<dim3((NB * 2 * CC + 255) / 256), 256, 0, stream>>>(ssq, NB * 2 * CC);

    // 3) depthwise 3x3 conv + fused q/k sum-of-squares
    dwconv_ssq<<<dim3(NN / 256, 3 * CC, NB), 256, 0, stream>>>(qkv, w_dw, qkvdw, ssq);

    // 4) attention: raw qk^T via WMMA, scale-by-norms + temperature, softmax
    attn_wmma<<<dim3(NHEADS, NB), 288, 0, stream>>>(qkvdw, ssq, temp, attn);

    // 5) fold attention into the output projection
    fuse_wout_attn<<<dim3((CC * CC) / 256, NB), 256, 0, stream>>>(w_out, attn, F);

    // 6) out = F_b (192x192, per-batch) x v_b (192x16384) -> final output
    gemm_wmma<<<dim3(NN / 64, CC / 32, NB), 256, 0, stream>>>(
        F, (long)CC * CC, qkvdw + (size_t)2 * CC * NN, (long)3 * CC * NN,
        out, (long)CC * NN);
}